// Prob_Imbalance_Loss_25177098289192
// MI455X (gfx1250) — compile-verified
//
#include <hip/hip_runtime.h>

typedef __attribute__((ext_vector_type(2))) float v2f;
typedef __attribute__((ext_vector_type(8))) float v8f;

#define NN 8192
#define ROWS_PER_WG 128
#define KCHUNK 64
#define KSEG 8
#define CHUNKS_PER_SEG ((NN / KSEG) / KCHUNK)  // 16
#define LDA 68             // A-tile LDS stride (conflict-free b64 frag loads)
#define LDP 40             // Pext-tile LDS stride (half-waves hit disjoint banks)
#define LDB 24             // per-block P rows stride
#define GW 32              // G row stride (21 rows x 32 cols)

// ---- gfx1250 async memory->LDS path (guarded; falls back to b128+ds_store) ----
#if defined(__has_builtin)
#if __has_builtin(__builtin_amdgcn_global_load_async_to_lds_b128)
#define HAVE_ASYNC 1
#endif
#endif
#ifndef HAVE_ASYNC
#define HAVE_ASYNC 0
#endif

#if HAVE_ASYNC
// Param 1 type per clang diagnostic: 'int __vector_size__(4*sizeof(int)) __device__ *'
// (__device__ == addrspace(1) in HIP diagnostics); param 2 is the LDS (as3) pointer.
typedef int v4i_g __attribute__((vector_size(16)));
typedef __attribute__((address_space(1))) v4i_g* g128_t;
typedef __attribute__((address_space(3))) v4i_g* l128_t;
#endif

__device__ __forceinline__ void wait_async0() {
#if HAVE_ASYNC
#if __has_builtin(__builtin_amdgcn_s_wait_asynccnt)
  __builtin_amdgcn_s_wait_asynccnt(0);
#else
  asm volatile("s_wait_asynccnt 0" ::: "memory");
#endif
#endif
}

__device__ __forceinline__ void stage_A(const float* __restrict__ A, float* dstLds,
                                        int row0, int kb, int tid) {
#pragma unroll
  for (int i = 0; i < 8; ++i) {
    const int j = tid + i * 256;
    const int r = j >> 4, c4 = j & 15;
    const float* gsrc = &A[(size_t)(row0 + r) * NN + kb + c4 * 4];
    float* ldst = &dstLds[r * LDA + c4 * 4];
#if HAVE_ASYNC
    __builtin_amdgcn_global_load_async_to_lds_b128((g128_t)gsrc, (l128_t)ldst, 0, 0);
#else
    *(float4*)ldst = *(const float4*)gsrc;
#endif
  }
}

__device__ __forceinline__ void stage_P(const float* __restrict__ P, float* dstLds,
                                        int kb, int tid) {
  // only the 64x20 live values; cols 20..31 are constant, written once up front
  for (int e = tid; e < KCHUNK * 20; e += 256) {
    const int r = e / 20, c = e - r * 20;
    dstLds[r * LDP + c] = P[(kb + r) * 20 + c];
  }
}

__global__ __launch_bounds__(256) void zero_G(float* G) {
  int e = blockIdx.x * 256 + threadIdx.x;
  if (e < 21 * GW) G[e] = 0.f;
}

// G = [P|1]^T * A * [P|1], accumulated across (row-block, k-segment) grid via atomics.
__global__ __launch_bounds__(256) void gemm_fused(const float* __restrict__ P,
                                                  const float* __restrict__ A,
                                                  float* __restrict__ G) {
  __shared__ float Alds[2][ROWS_PER_WG * LDA];  // 2 x 34.0 KB (double buffer)
  __shared__ float Plds[2][KCHUNK * LDP];       // 2 x 10.0 KB (double buffer)
  __shared__ float Pblk[ROWS_PER_WG * LDB];     // 12.0 KB
  __shared__ float Wlds[21 * GW];               //  2.6 KB

  const int tid  = threadIdx.x;
  const int lane = tid & 31;
  const int wave = tid >> 5;          // 8 waves, each owns a 16-row tile
  const int half = lane >> 4;
  const int l15  = lane & 15;
  const int row0   = blockIdx.x * ROWS_PER_WG;
  const int kstart = blockIdx.y * (NN / KSEG);

  // One-time init: Pext constant columns (ones at 20, zeros 21..31) in both buffers.
  for (int e = tid; e < 2 * KCHUNK; e += 256) {
    float* row = &Plds[e >> 6][(e & 63) * LDP];
    row[20] = 1.0f;
#pragma unroll
    for (int c = 21; c < 32; ++c) row[c] = 0.f;
  }
  // This block's Pext rows (col 20 = 1.0) for the epilogue contraction.
  for (int e = tid; e < ROWS_PER_WG * 21; e += 256) {
    int r = e / 21, c = e % 21;
    Pblk[r * LDB + c] = (c < 20) ? P[(row0 + r) * 20 + c] : 1.0f;
  }
  for (int e = tid; e < 21 * GW; e += 256) Wlds[e] = 0.f;

  // Prologue: stage chunk 0 into buffer 0.
  stage_A(A, &Alds[0][0], row0, kstart, tid);
  stage_P(P, &Plds[0][0], kstart, tid);
  wait_async0();
  __syncthreads();

  v8f acc0 = {};  // AP tile, cols 0..15
  v8f acc1 = {};  // AP tile, cols 16..31

  for (int ch = 0; ch < CHUNKS_PER_SEG; ++ch) {
    const int cur = ch & 1;
    const int kb  = kstart + ch * KCHUNK;
    // Kick off next chunk's staging into the other buffer (overlaps with WMMAs).
    if (ch + 1 < CHUNKS_PER_SEG) {
      stage_A(A, &Alds[cur ^ 1][0], row0, kb + KCHUNK, tid);
      stage_P(P, &Plds[cur ^ 1][0], kb + KCHUNK, tid);
    }
    const float* __restrict__ Ab = &Alds[cur][0];
    const float* __restrict__ Pb = &Plds[cur][0];
#pragma unroll
    for (int kk = 0; kk < KCHUNK; kk += 4) {
      // A frag (16x4 f32): lanes 0-15 hold K=kk..kk+1, lanes 16-31 K=kk+2..kk+3.
      v2f a = *(const v2f*)&Ab[(wave * 16 + l15) * LDA + kk + 2 * half];
      const int krow = kk + 2 * half;
      v2f b0, b1;
      b0.x = Pb[krow * LDP + l15];
      b0.y = Pb[(krow + 1) * LDP + l15];
      acc0 = __builtin_amdgcn_wmma_f32_16x16x4_f32(false, a, false, b0,
                                                   (short)0, acc0, false, false);
      b1.x = Pb[krow * LDP + 16 + l15];
      b1.y = Pb[(krow + 1) * LDP + 16 + l15];
      acc1 = __builtin_amdgcn_wmma_f32_16x16x4_f32(false, a, false, b1,
                                                   (short)0, acc1, false, false);
    }
    wait_async0();     // next chunk's async tile fully landed in LDS
    __syncthreads();   // visible to all waves; safe to overwrite read buffer
  }
  __syncthreads();

  // Epilogue: Wlds[k][n] += Pext[m][k] * AP[m][n]  (LDS float atomics).
  // C/D layout: VGPR v -> M = v (lanes 0-15) or v+8 (lanes 16-31), N = l15 (+16*tile).
#pragma unroll
  for (int v = 0; v < 8; ++v) {
    const int mloc = wave * 16 + v + 8 * half;
    const float* prow = &Pblk[mloc * LDB];
    const float ap0 = acc0[v];
    const float ap1 = acc1[v];
#pragma unroll
    for (int k = 0; k <= 20; ++k) {
      const float w = prow[k];
      atomicAdd(&Wlds[k * GW + l15],      w * ap0);
      atomicAdd(&Wlds[k * GW + 16 + l15], w * ap1);
    }
  }
  __syncthreads();
  for (int e = tid; e < 21 * GW; e += 256) atomicAdd(&G[e], Wlds[e]);
}

__global__ __launch_bounds__(64) void finalize(const float* __restrict__ G,
                                               const int* __restrict__ selp,
                                               float* __restrict__ out) {
  __shared__ float W[20][20];
  __shared__ float vol[20];
  __shared__ float pv[190];
  const int tid = threadIdx.x;  // 64 threads
  for (int e = tid; e < 400; e += 64) W[e / 20][e % 20] = G[(e / 20) * GW + (e % 20)];
  if (tid < 20) vol[tid] = G[tid * GW + 20] + G[20 * GW + tid];
  __syncthreads();
  for (int e = tid; e < 190; e += 64) {
    int k = 0, rem = e;
    while (rem >= 19 - k) { rem -= 19 - k; ++k; }
    int l = k + 1 + rem;
    float d = fabsf(W[k][l] - W[l][k]);
    pv[e] = (d != 0.f) ? (d / (vol[k] + vol[l] + 1e-10f) * 2.f) : 0.f;
  }
  __syncthreads();
  if (tid == 0) {
    const int sel = selp[0];
    float s = 0.f;
    for (int it = 0; it < sel; ++it) {
      int bi = 0; float bv = pv[0];
      for (int e = 1; e < 190; ++e)
        if (pv[e] > bv) { bv = pv[e]; bi = e; }
      s += bv;
      pv[bi] = -1.f;
    }
    out[0] = 1.f - s / (float)sel;
  }
}

extern "C" void kernel_launch(void* const* d_in, const int* in_sizes, int n_in,
                              void* d_out, int out_size, void* d_ws, size_t ws_size,
                              hipStream_t stream) {
  (void)in_sizes; (void)n_in; (void)out_size; (void)ws_size;
  const float* P   = (const float*)d_in[0];
  const float* A   = (const float*)d_in[1];
  const int*  selp = (const int*)d_in[3];
  float* out = (float*)d_out;
  float* G   = (float*)d_ws;  // 21*32 floats = 2688 bytes

  zero_G<<<dim3((21 * GW + 255) / 256), 256, 0, stream>>>(G);
  gemm_fused<<<dim3(NN / ROWS_PER_WG, KSEG), 256, 0, stream>>>(P, A, G);
  finalize<<<1, 64, 0, stream>>>(G, selp, out);
}